// ApplyKernels_29137058136307
// MI455X (gfx1250) — compile-verified
//
#include <hip/hip_runtime.h>
#include <stdint.h>

#define KS      5
#define W       508           // output H = W
#define HIN     512           // input tensor H = W
#define NCH     3
#define NB      8
#define TX      64            // tile width  (x)
#define TY      16            // tile height (y)
#define HALO    (KS - 1)      // 4
#define LDS_W   (TX + HALO)   // 68 floats per staged row (= 17 x b128)
#define LDS_H   (TY + HALO)   // 20 staged rows
#define NSEG    (LDS_W / 4)   // 17 b128 segments per row
#define NTRANS  (NCH * LDS_H * NSEG)  // 1020 async transfers per block

__global__ __launch_bounds__(256) void apply_kernels_gfx1250(
    const float* __restrict__ kern,    // (8, 25, 508, 508)
    const float* __restrict__ tens,    // (8,  3, 512, 512)
    float* __restrict__ out_w,         // (8,  3, 508, 508)
    float* __restrict__ out_s)         // (8,  1, 508, 508)
{
    __shared__ float smem[NCH * LDS_H * LDS_W];

    const int tid = threadIdx.x;
    const int x0  = blockIdx.x * TX;
    const int y0  = blockIdx.y * TY;
    const int b   = blockIdx.z;

    // ---------------- async stage: tensor halo tile -> LDS -----------------
    // 3 channels x 20 rows x 17 b128 segments = 1020 transfers, <=4 per lane,
    // issued back-to-back on the async queue (tracked by ASYNCcnt).
    const uint32_t lds_base = (uint32_t)(uintptr_t)&smem[0];
#pragma unroll
    for (int i = 0; i < 4; ++i) {
        const int w = tid + i * 256;
        if (w < NTRANS) {
            const int seg = w % NSEG;
            const int rc  = w / NSEG;
            const int r   = rc % LDS_H;
            const int c   = rc / LDS_H;
            int yy = y0 + r;       if (yy > HIN - 1) yy = HIN - 1;  // edge clamp
            int xx = x0 + seg * 4; if (xx > HIN - 4) xx = HIN - 4;  // keep b128 in-bounds
            const float* gsrc =
                tens + ((size_t)(b * NCH + c) * HIN + (size_t)yy) * HIN + xx;
            const uint32_t ldso = lds_base +
                (uint32_t)(((c * LDS_H + r) * LDS_W + seg * 4) * 4);
            asm volatile("global_load_async_to_lds_b128 %0, %1, off"
                         :: "v"(ldso), "v"((uint64_t)(uintptr_t)gsrc)
                         : "memory");
        }
    }

    // ---------------- per-thread output mapping ----------------------------
    const int tx   = tid & (TX - 1);   // 0..63
    const int tg   = tid >> 6;         // 0..3
    const int row0 = tg * 4;           // 4 consecutive output rows per thread
    const int ox   = x0 + tx;

    float a0[4], a1[4], a2[4], ks[4];
    size_t koff[4];
    const float* kb = kern + (size_t)b * (KS * KS) * W * W;
#pragma unroll
    for (int j = 0; j < 4; ++j) {
        a0[j] = a1[j] = a2[j] = ks[j] = 0.0f;
        int oy  = y0 + row0 + j; if (oy > W - 1) oy = W - 1;   // clamped addr;
        int oxc = ox;            if (oxc > W - 1) oxc = W - 1; // result unused OOB
        koff[j] = (size_t)oy * W + oxc;
        // Warm the dominant HBM stream while the async staging is in flight:
        // pulls the first tap rows of this thread's kernel stream toward L2/L0.
        __builtin_prefetch(kb + koff[j], 0, 0);                // tap 0
        __builtin_prefetch(kb + (size_t)W * W + koff[j], 0, 0);// tap 1
    }

    asm volatile("s_wait_asynccnt 0" ::: "memory");
    __syncthreads();

    // ---------------- per-pixel 25-tap spatially-varying filter ------------
#pragma unroll 5
    for (int t = 0; t < KS * KS; ++t) {
        const int di = t / KS;
        const int dj = t - di * KS;
        const float* kp = kb + (size_t)t * W * W;
#pragma unroll
        for (int j = 0; j < 4; ++j) {
            const float k = kp[koff[j]];           // coalesced HBM stream
            ks[j] += k;
            const float* sp = &smem[(row0 + j + di) * LDS_W + (tx + dj)];
            a0[j] += k * sp[0 * LDS_H * LDS_W];
            a1[j] += k * sp[1 * LDS_H * LDS_W];
            a2[j] += k * sp[2 * LDS_H * LDS_W];
        }
    }

    // ---------------- store -------------------------------------------------
    if (ox < W) {
        const size_t pw = (size_t)W * W;
#pragma unroll
        for (int j = 0; j < 4; ++j) {
            const int oy = y0 + row0 + j;
            if (oy < W) {
                const size_t o = (size_t)oy * W + ox;
                out_w[((size_t)(b * NCH) + 0) * pw + o] = a0[j];
                out_w[((size_t)(b * NCH) + 1) * pw + o] = a1[j];
                out_w[((size_t)(b * NCH) + 2) * pw + o] = a2[j];
                out_s[(size_t)b * pw + o]               = ks[j];
            }
        }
    }
}

extern "C" void kernel_launch(void* const* d_in, const int* in_sizes, int n_in,
                              void* d_out, int out_size, void* d_ws, size_t ws_size,
                              hipStream_t stream) {
    (void)in_sizes; (void)n_in; (void)d_ws; (void)ws_size; (void)out_size;
    const float* kern = (const float*)d_in[0];  // (8,25,508,508) f32
    const float* tens = (const float*)d_in[1];  // (8, 3,512,512) f32
    float* out   = (float*)d_out;
    float* out_w = out;                                   // (8,3,508,508)
    float* out_s = out + (size_t)NB * NCH * W * W;        // (8,1,508,508)

    dim3 grid((W + TX - 1) / TX,   // 8
              (W + TY - 1) / TY,   // 32
              NB);                 // 8
    apply_kernels_gfx1250<<<grid, 256, 0, stream>>>(kern, tens, out_w, out_s);
}